// GraphLayerLinearGraphGen_51101520888315
// MI455X (gfx1250) — compile-verified
//
#include <hip/hip_runtime.h>
#include <math.h>

// ---------------------------------------------------------------------------
// Problem constants (reference: N=64, K=64, C_IN=128, C=128)
//   rows M = N*K = 4096, inner/outer channel dim = 128 everywhere.
// ---------------------------------------------------------------------------
#define MROWS 4096
#define CDIM  128

typedef __attribute__((ext_vector_type(2))) float v2f;
typedef __attribute__((ext_vector_type(8))) float v8f;

// V_WMMA_F32_16X16X4_F32 : D(16x16 f32) = A(16x4 f32) * B(4x16 f32) + C
__device__ __forceinline__ v8f wmma_f32(v2f a, v2f b, v8f c) {
    return __builtin_amdgcn_wmma_f32_16x16x4_f32(
        /*neg_a=*/false, a, /*neg_b=*/false, b,
        /*c_mod=*/(short)0, c, /*reuse_a=*/false, /*reuse_b=*/false);
}

// ---------------------------------------------------------------------------
// Generic fp32 WMMA GEMM:  Y[M x 128] = X1[M x 128] * W1^T (+ X2 * W2^T)
//   + bias1 + bias2_scale*bias2 + t * W1[:, tcol]      , optional tanh.
// W matrices are row-major [128 x ldw] (row = output channel).
// Each wave computes a 16x32 tile (two 16x16 accumulators, A reused).
// Block = 128 threads (4 waves) -> 16 rows x full 128 cols. Grid = M/16.
// ---------------------------------------------------------------------------
__global__ __launch_bounds__(128) void gemm_wmma_f32_kernel(
    const float* __restrict__ X1, const float* __restrict__ W1, int ldw1,
    const float* __restrict__ X2, const float* __restrict__ W2, int ldw2,
    const float* __restrict__ bias1, const float* __restrict__ bias2,
    float bias2_scale,
    const float* __restrict__ tptr, int tcol,
    float* __restrict__ Y, int do_tanh)
{
    const int lane = threadIdx.x & 31;
    const int wv   = threadIdx.x >> 5;
    const int m0   = blockIdx.x * 16;      // output row tile
    const int n0   = wv * 32;              // output col tile (per wave)
    const int rA   = lane & 15;            // A-row / B-col within tile
    const int kh   = (lane >> 4) << 1;     // k sub-offset: 0 or 2

    v8f acc0 = {};
    v8f acc1 = {};

    // ---- pass 1: X1 * W1^T ----
    {
        const float* xr  = X1 + (size_t)(m0 + rA) * CDIM;
        const float* wr0 = W1 + (size_t)(n0 + rA) * ldw1;
        const float* wr1 = W1 + (size_t)(n0 + 16 + rA) * ldw1;
#pragma unroll 8
        for (int k = 0; k < CDIM; k += 4) {
            v2f a, b0, b1;
            a.x  = xr [k + kh];  a.y  = xr [k + kh + 1];
            b0.x = wr0[k + kh];  b0.y = wr0[k + kh + 1];
            b1.x = wr1[k + kh];  b1.y = wr1[k + kh + 1];
            acc0 = wmma_f32(a, b0, acc0);
            acc1 = wmma_f32(a, b1, acc1);
        }
    }
    // ---- optional pass 2: + X2 * W2^T (same accumulators) ----
    if (X2 != nullptr) {
        const float* xr  = X2 + (size_t)(m0 + rA) * CDIM;
        const float* wr0 = W2 + (size_t)(n0 + rA) * ldw2;
        const float* wr1 = W2 + (size_t)(n0 + 16 + rA) * ldw2;
#pragma unroll 8
        for (int k = 0; k < CDIM; k += 4) {
            v2f a, b0, b1;
            a.x  = xr [k + kh];  a.y  = xr [k + kh + 1];
            b0.x = wr0[k + kh];  b0.y = wr0[k + kh + 1];
            b1.x = wr1[k + kh];  b1.y = wr1[k + kh + 1];
            acc0 = wmma_f32(a, b0, acc0);
            acc1 = wmma_f32(a, b1, acc1);
        }
    }

    // ---- epilogue: D layout -> VGPR rr holds rows (rr, rr+8), col = lane%16
    const int   mhi  = (lane >> 4) * 8;
    const float tval = tptr ? tptr[0] : 0.0f;
#pragma unroll
    for (int tacc = 0; tacc < 2; ++tacc) {
        const int nOut = n0 + tacc * 16 + rA;
        float bc = 0.0f;
        if (bias1) bc += bias1[nOut];
        if (tptr)  bc += tval * W1[(size_t)nOut * ldw1 + tcol];
        if (bias2) bc += bias2_scale * bias2[nOut];
        v8f acc = tacc ? acc1 : acc0;
#pragma unroll
        for (int rr = 0; rr < 8; ++rr) {
            float v = acc[rr] + bc;
            if (do_tanh) v = tanhf(v);
            Y[(size_t)(m0 + rr + mhi) * CDIM + nOut] = v;
        }
    }
}

// ---------------------------------------------------------------------------
// Binary-message reduction (the Wb2 GEMM is hoisted out of the i-sum):
//   Ss[n,j,c] = ( sum_{i=0..63} tanh(Apre[n,j,c] + B[n,i,c])
//                 - tanh(Apre[n,j,c] + B[n,j,c]) ) / 64
// Apre already contains bb1 (folded into the A GEMM bias).
// Grid (64, 8), 256 threads. B[n] (32 KB) staged in LDS.
// ---------------------------------------------------------------------------
__global__ __launch_bounds__(256) void binary_reduce_kernel(
    const float* __restrict__ Apre, const float* __restrict__ Bbuf,
    float* __restrict__ Ss)
{
    __shared__ float bs[64 * CDIM];   // 32 KB of the 320 KB WGP LDS
    const int n = blockIdx.x;
    const float* Bn = Bbuf + (size_t)n * 64 * CDIM;
    for (int idx = threadIdx.x; idx < 64 * CDIM; idx += 256) bs[idx] = Bn[idx];
    __syncthreads();

    const int jbase = blockIdx.y * 8;                 // 8 j's per block
    for (int e = threadIdx.x; e < 8 * CDIM; e += 256) {
        const int j = jbase + (e >> 7);
        const int c = e & (CDIM - 1);
        const float av = Apre[((size_t)n * 64 + j) * CDIM + c];
        float s = 0.0f;
#pragma unroll 4
        for (int i = 0; i < 64; ++i) s += tanhf(av + bs[i * CDIM + c]);
        s -= tanhf(av + bs[j * CDIM + c]);            // remove i == j term
        Ss[((size_t)n * 64 + j) * CDIM + c] = s * (1.0f / 64.0f);
    }
}

// ---------------------------------------------------------------------------
// Launch
// ---------------------------------------------------------------------------
extern "C" void kernel_launch(void* const* d_in, const int* in_sizes, int n_in,
                              void* d_out, int out_size, void* d_ws, size_t ws_size,
                              hipStream_t stream)
{
    (void)in_sizes; (void)n_in; (void)out_size; (void)ws_size;

    const float* t   = (const float*)d_in[0];   // [1]
    const float* x   = (const float*)d_in[1];   // [64,64,128]
    const float* Wl  = (const float*)d_in[2];   // [128,129]
    const float* bl  = (const float*)d_in[3];   // [128]
    const float* Wu1 = (const float*)d_in[4];   // [128,128]
    const float* bu1 = (const float*)d_in[5];
    const float* Wu2 = (const float*)d_in[6];
    const float* bu2 = (const float*)d_in[7];
    const float* Wb1 = (const float*)d_in[8];   // [128,256]
    const float* bb1 = (const float*)d_in[9];
    const float* Wb2 = (const float*)d_in[10];  // [128,128]
    const float* bb2 = (const float*)d_in[11];
    float* out = (float*)d_out;                 // [64,64,128]

    // scratch layout (5 x 2 MB = 10 MB)
    float* h  = (float*)d_ws;
    float* Ab = h  + (size_t)MROWS * CDIM;
    float* Bb = Ab + (size_t)MROWS * CDIM;
    float* T1 = Bb + (size_t)MROWS * CDIM;
    float* Ss = T1 + (size_t)MROWS * CDIM;

    const dim3 gg(MROWS / 16), gb(128);

    // 1) h = x @ Wl[:, :128]^T + t*Wl[:,128] + bl     (ldw = 129, t-column 128)
    gemm_wmma_f32_kernel<<<gg, gb, 0, stream>>>(
        x, Wl, 129, nullptr, nullptr, 0, bl, nullptr, 0.0f, t, 128, h, 0);

    // 2) Apre = h @ Wb1[:, :128]^T + bb1              (bb1 folded here)
    gemm_wmma_f32_kernel<<<gg, gb, 0, stream>>>(
        h, Wb1, 256, nullptr, nullptr, 0, bb1, nullptr, 0.0f, nullptr, 0, Ab, 0);

    // 3) B    = h @ Wb1[:, 128:]^T
    gemm_wmma_f32_kernel<<<gg, gb, 0, stream>>>(
        h, Wb1 + 128, 256, nullptr, nullptr, 0, nullptr, nullptr, 0.0f, nullptr, 0, Bb, 0);

    // 4) T1   = tanh(h @ Wu1^T + bu1)
    gemm_wmma_f32_kernel<<<gg, gb, 0, stream>>>(
        h, Wu1, 128, nullptr, nullptr, 0, bu1, nullptr, 0.0f, nullptr, 0, T1, 1);

    // 5) Ss[n,j] = (sum_{i!=j} tanh(Apre[n,j] + B[n,i])) / 64
    binary_reduce_kernel<<<dim3(64, 8), dim3(256), 0, stream>>>(Ab, Bb, Ss);

    // 6) out = T1 @ Wu2^T + Ss @ Wb2^T + bu2 + (63/64)*bb2
    gemm_wmma_f32_kernel<<<gg, gb, 0, stream>>>(
        T1, Wu2, 128, Ss, Wb2, 128, bu2, bb2, 63.0f / 64.0f, nullptr, 0, out, 0);
}